// GINModel_72619307041225
// MI455X (gfx1250) — compile-verified
//
#include <hip/hip_runtime.h>

#define N_NODES 100000
#define N_EDGES 1600000
#define D 64

// Padded LDS strides (words) for conflict-free access on 64 banks:
#define SA_STRIDE 68    // A tile row stride: banks kk+4m(+0..3) -> all 64 banks distinct
#define SW_STRIDE 160   // W pair-row stride: 160 % 64 == 32 -> lane halves use disjoint bank sets

typedef __attribute__((ext_vector_type(2))) float v2f;
typedef __attribute__((ext_vector_type(8))) float v8f;

// Order-preserving float->uint encoding: enc(a) < enc(b) <=> a < b (IEEE total order incl. -0<+0)
__device__ __forceinline__ unsigned enc_f32(float f) {
    unsigned u = __float_as_uint(f);
    return (u & 0x80000000u) ? ~u : (u | 0x80000000u);
}
__device__ __forceinline__ float dec_f32(unsigned u) {
    return (u & 0x80000000u) ? __uint_as_float(u ^ 0x80000000u) : __uint_as_float(~u);
}
#define ENC_NEG_INF 0x007FFFFFu  // enc(-inf): identity of max, sentinel for "no in-edges"

// ---------------------------------------------------------------------------
// Kernel 1: init aggregation buffer to enc(-inf)
// ---------------------------------------------------------------------------
__global__ __launch_bounds__(256) void gin_init_agg(unsigned* __restrict__ agg, int n) {
    int i = blockIdx.x * 256 + threadIdx.x;
    if (i < n) agg[i] = ENC_NEG_INF;
}

// ---------------------------------------------------------------------------
// Kernel 2: edge scatter-max. thread = (edge, 4-feature chunk).
// 16 consecutive lanes cover one 256B source row -> coalesced gather,
// 4x GLOBAL_ATOMIC_MAX_U32 per thread (single-shot HW atomic, L2-resident:
// node table 25.6MB + agg 25.6MB both fit the 192MB L2).
// ---------------------------------------------------------------------------
__global__ __launch_bounds__(256) void gin_scatter_max(const float* __restrict__ x,
                                                       const int* __restrict__ src,
                                                       const int* __restrict__ dst,
                                                       unsigned* __restrict__ agg) {
    int idx = blockIdx.x * 256 + threadIdx.x;   // N_EDGES*16 threads
    int e  = idx >> 4;
    int c4 = (idx & 15) << 2;
    if (e >= N_EDGES) return;
    int s = src[e];
    int d = dst[e];
    const float4 v = *(const float4*)(x + (size_t)s * D + c4);
    unsigned* a = agg + (size_t)d * D + c4;
    atomicMax(a + 0, enc_f32(v.x));
    atomicMax(a + 1, enc_f32(v.y));
    atomicMax(a + 2, enc_f32(v.z));
    atomicMax(a + 3, enc_f32(v.w));
}

// ---------------------------------------------------------------------------
// Kernel 3: fused rst = x + where(agg==-inf,0,agg); out = rst @ W^T + b
// One wave -> 16 nodes x 64 outputs via 4 accumulators of V_WMMA_F32_16X16X4_F32.
// Block = 256 threads = 8 waves = 128 nodes.
// B fragments come from a pair-interleaved W layout so each operand is a
// single contiguous ds_load_b64 into an even-aligned VGPR pair (no packing movs).
// ---------------------------------------------------------------------------
__global__ __launch_bounds__(256) void gin_wmma_gemm(const float* __restrict__ x,
                                                     const unsigned* __restrict__ agg,
                                                     const float* __restrict__ W,
                                                     const float* __restrict__ bias,
                                                     float* __restrict__ out) {
    // sWp[(k>>1)*SW_STRIDE + n*2 + (k&1)] = W[n][k]   (B matrix pairs along K)
    __shared__ float sWp[(D / 2) * SW_STRIDE];            // 20 KB
    __shared__ float sA[8][16 * SA_STRIDE];               // 34 KB, per-wave rst tiles

    const int tid  = threadIdx.x;
    const int wave = tid >> 5;
    const int lane = tid & 31;
    const int node_base = blockIdx.x * 128 + wave * 16;

    // --- Stage W pair-interleaved into LDS (coalesced global reads of W) ---
#pragma unroll
    for (int i = 0; i < 16; ++i) {
        int idx = tid + i * 256;       // 4096 elements
        int o = idx >> 6;              // output dim n (row of W)
        int k = idx & 63;              // input dim  k (col of W)
        sWp[(k >> 1) * SW_STRIDE + o * 2 + (k & 1)] = W[idx];
    }

    // --- Stage rst tile into LDS: coalesced float4 loads, clamp tail rows ---
#pragma unroll
    for (int j = 0; j < 8; ++j) {
        int p   = lane + j * 32;       // 256 float4 chunks cover 16x64 tile
        int row = p >> 4;
        int c4  = (p & 15) << 2;
        int g   = node_base + row;
        if (g >= N_NODES) g = N_NODES - 1;
        const float4 xin = *(const float4*)(x   + (size_t)g * D + c4);
        const uint4  au  = *(const uint4*) (agg + (size_t)g * D + c4);
        float4 r;
        r.x = xin.x + (au.x == ENC_NEG_INF ? 0.0f : dec_f32(au.x));
        r.y = xin.y + (au.y == ENC_NEG_INF ? 0.0f : dec_f32(au.y));
        r.z = xin.z + (au.z == ENC_NEG_INF ? 0.0f : dec_f32(au.z));
        r.w = xin.w + (au.w == ENC_NEG_INF ? 0.0f : dec_f32(au.w));
        *(float4*)(&sA[wave][row * SA_STRIDE + c4]) = r;   // 272B row stride keeps 16B align
    }
    __syncthreads();

    // --- WMMA accumulation: C(16x16) += A(16x4) x B(4x16), K = 0..60 step 4 ---
    // A frag (ISA 16x4 f32 layout): lane<16 holds (M=lane, K=k..k+1),
    //                               lane>=16 holds (M=lane-16, K=k+2..k+3)
    // B frag (4x16): vgpr pair = rows {kk, kk+1} striped over lane halves
    v8f acc[4];
    acc[0] = v8f{}; acc[1] = v8f{}; acc[2] = v8f{}; acc[3] = v8f{};

    const int m     = lane & 15;
    const int khalf = (lane >> 4) << 1;        // 0 for lanes 0-15, 2 for lanes 16-31
    const float* Arow = &sA[wave][m * SA_STRIDE];

#pragma unroll
    for (int k = 0; k < D; k += 4) {
        const int kk = k + khalf;              // even -> 8B-aligned v2f loads
        const v2f a = *(const v2f*)(Arow + kk);
        const float* bbase = &sWp[(kk >> 1) * SW_STRIDE + m * 2];
#pragma unroll
        for (int t = 0; t < 4; ++t) {
            const v2f bf = *(const v2f*)(bbase + t * 32);  // {W[n][kk], W[n][kk+1]}
            acc[t] = __builtin_amdgcn_wmma_f32_16x16x4_f32(
                false, a, false, bf, (short)0, acc[t], false, false);
        }
    }

    // --- Store C tiles + bias; predicate only the stores on the tail block ---
    const int rowoff = (lane >> 4) << 3;       // lane>=16 -> rows 8..15
#pragma unroll
    for (int t = 0; t < 4; ++t) {
        const int col = t * 16 + m;
        const float bv = bias[col];
#pragma unroll
        for (int v = 0; v < 8; ++v) {
            const int row = node_base + rowoff + v;
            if (row < N_NODES) out[(size_t)row * D + col] = acc[t][v] + bv;
        }
    }
}

// ---------------------------------------------------------------------------
extern "C" void kernel_launch(void* const* d_in, const int* in_sizes, int n_in,
                              void* d_out, int out_size, void* d_ws, size_t ws_size,
                              hipStream_t stream) {
    const float* x    = (const float*)d_in[0];
    const float* W    = (const float*)d_in[1];
    const float* bias = (const float*)d_in[2];
    const int*   src  = (const int*)d_in[3];
    const int*   dst  = (const int*)d_in[4];
    float*       out  = (float*)d_out;
    unsigned*    agg  = (unsigned*)d_ws;       // N_NODES*64 u32 = 25.6 MB scratch

    const int total = N_NODES * D;
    gin_init_agg<<<(total + 255) / 256, 256, 0, stream>>>(agg, total);

    const int sc_threads = N_EDGES * 16;
    gin_scatter_max<<<(sc_threads + 255) / 256, 256, 0, stream>>>(x, src, dst, agg);

    gin_wmma_gemm<<<(N_NODES + 127) / 128, 256, 0, stream>>>(x, agg, W, bias, out);
}